// MeanIoU_55130200211621
// MI455X (gfx1250) — compile-verified
//
#include <hip/hip_runtime.h>

#define NUM_CLASSES 21
#define THREADS     256
#define TILE_PIX    256
#define ROW_F       NUM_CLASSES            // floats per pixel (21)
#define TILE_F      (TILE_PIX * ROW_F)     // 5376 floats per tensor tile
#define TILE_VEC    (TILE_F / 4)           // 1344 b128 chunks (exact: 21504 B / 16)

#if defined(__HIP_DEVICE_COMPILE__)
typedef int v4i __attribute__((ext_vector_type(4)));
typedef __attribute__((address_space(1))) v4i* gbl_v4i_ptr;
typedef __attribute__((address_space(3))) v4i* lds_v4i_ptr;

// ---- CDNA5 async global->LDS copy (ASYNCcnt-tracked), 16 bytes per lane ----
__device__ __forceinline__ void async_copy_b128(const void* gsrc, void* ldst) {
#if __has_builtin(__builtin_amdgcn_global_load_async_to_lds_b128)
  __builtin_amdgcn_global_load_async_to_lds_b128(
      (gbl_v4i_ptr)(v4i*)const_cast<void*>(gsrc),
      (lds_v4i_ptr)(v4i*)ldst,
      /*offset=*/0, /*cpol=*/0);
#else
  unsigned laddr = (unsigned)(unsigned long long)
      (__attribute__((address_space(3))) char*)(char*)ldst;
  asm volatile("global_load_async_to_lds_b128 %0, %1, off"
               :: "v"(laddr), "v"(gsrc) : "memory");
#endif
}

__device__ __forceinline__ void wait_asynccnt0() {
#if __has_builtin(__builtin_amdgcn_s_wait_asynccnt)
  __builtin_amdgcn_s_wait_asynccnt(0);
#else
  asm volatile("s_wait_asynccnt 0" ::: "memory");
#endif
}
#endif // __HIP_DEVICE_COMPILE__

// ---------------- kernel 0: zero the 63-bin global histogram ----------------
__global__ void miou_zero_kernel(unsigned* g) {
#if defined(__HIP_DEVICE_COMPILE__)
  if (threadIdx.x < 3 * NUM_CLASSES) g[threadIdx.x] = 0u;
#endif
}

// ---- kernel 1: tile -> LDS (async), per-pixel dual argmax, LDS histogram ----
__global__ __launch_bounds__(THREADS)
void miou_hist_kernel(const float* __restrict__ yt,
                      const float* __restrict__ yp,
                      unsigned* __restrict__ ghist,
                      long long pixels) {
#if defined(__HIP_DEVICE_COMPILE__)
  __shared__ float    lt[TILE_F];
  __shared__ float    lp[TILE_F];
  __shared__ unsigned hist[3 * NUM_CLASSES];

  const int tid = threadIdx.x;
  if (tid < 3 * NUM_CLASSES) hist[tid] = 0u;

  const long long pix0 = (long long)blockIdx.x * TILE_PIX;
  const float* gt = yt + pix0 * ROW_F;
  const float* gp = yp + pix0 * ROW_F;
  const bool full = (pix0 + TILE_PIX) <= pixels;

  if (full) {
    // 16B-aligned (256*84 = 1344*16), fully coalesced async b128 stream.
    for (int i = tid; i < TILE_VEC; i += THREADS) {
      async_copy_b128(gt + i * 4, &lt[i * 4]);
      async_copy_b128(gp + i * 4, &lp[i * 4]);
    }
    wait_asynccnt0();
  } else {
    // Ragged tail tile (not hit for the reference shape): scalar staging.
    const int nf = (int)((pixels - pix0) * ROW_F);
    for (int i = tid; i < nf; i += THREADS) { lt[i] = gt[i]; lp[i] = gp[i]; }
  }
  __syncthreads();

  const long long pix = pix0 + tid;
  if (pix < pixels) {
    // Row stride 21 dwords (odd) -> conflict-free across 64 LDS banks.
    const float* rt = &lt[tid * ROW_F];
    const float* rp = &lp[tid * ROW_F];
    int ta = 0, pa = 0;
    float tm = rt[0], pm = rp[0];
#pragma unroll
    for (int c = 1; c < NUM_CLASSES; ++c) {
      float tv = rt[c]; if (tv > tm) { tm = tv; ta = c; }  // first-max wins (argmax semantics)
      float pv = rp[c]; if (pv > pm) { pm = pv; pa = c; }
    }
    atomicAdd(&hist[ta], 1u);                                 // t_cnt
    atomicAdd(&hist[NUM_CLASSES + pa], 1u);                   // p_cnt
    if (ta == pa) atomicAdd(&hist[2 * NUM_CLASSES + ta], 1u); // intersection
  }
  __syncthreads();

  if (tid < 3 * NUM_CLASSES) {
    unsigned v = hist[tid];
    if (v) atomicAdd(&ghist[tid], v);
  }
#endif
}

// ---------------- kernel 2: 21-class IoU reduction -> scalar ----------------
__global__ void miou_final_kernel(const unsigned* __restrict__ ghist,
                                  float* __restrict__ out) {
#if defined(__HIP_DEVICE_COMPILE__)
  if (threadIdx.x == 0 && blockIdx.x == 0) {
    float s = 0.0f, nv = 0.0f;
#pragma unroll
    for (int c = 0; c < NUM_CLASSES; ++c) {
      float t  = (float)ghist[c];
      float p  = (float)ghist[NUM_CLASSES + c];
      float in = (float)ghist[2 * NUM_CLASSES + c];
      float un = t + p - in;
      if (un > 0.0f) { s += in / un; nv += 1.0f; }
    }
    out[0] = (nv > 0.0f) ? (s / nv) : 0.0f;
  }
#endif
}

extern "C" void kernel_launch(void* const* d_in, const int* in_sizes, int n_in,
                              void* d_out, int out_size, void* d_ws, size_t ws_size,
                              hipStream_t stream) {
  const float* yt = (const float*)d_in[0];
  const float* yp = (const float*)d_in[1];
  unsigned*  hist = (unsigned*)d_ws;     // 63 bins in scratch
  float*      out = (float*)d_out;

  const long long nelem  = (long long)in_sizes[0];        // 8*512*512*21
  const long long pixels = nelem / NUM_CLASSES;           // 2,097,152
  const int blocks = (int)((pixels + TILE_PIX - 1) / TILE_PIX);  // 8192

  miou_zero_kernel<<<1, 64, 0, stream>>>(hist);
  miou_hist_kernel<<<blocks, THREADS, 0, stream>>>(yt, yp, hist, pixels);
  miou_final_kernel<<<1, 32, 0, stream>>>(hist, out);
}